// StrokeRenderer_42614665511078
// MI455X (gfx1250) — compile-verified
//
#include <hip/hip_runtime.h>
#include <math.h>

typedef __attribute__((ext_vector_type(2))) float v2f;
typedef __attribute__((ext_vector_type(8))) float v8f;

#define GSZ 256
#define NSEG 960      // 64 strokes * 15 segments
#define NGRP 60       // 960 / 16 segments per WMMA tile
#define THREADS 256

__global__ __launch_bounds__(THREADS)
void stroke_render_wmma(const float* __restrict__ strokes, float* __restrict__ out) {
    // Per-segment raw data, shared by all waves in the block.
    __shared__ float4 segA[NSEG];  // vx, vy, sx, sy   (pixel space)
    __shared__ float2 segB[NSEG];  // d2, -0.5/(d2+eps)

    const int tid = threadIdx.x;
    for (int s = tid; s < NSEG; s += THREADS) {
        int st = s / 15, k = s - st * 15;
        const float* p0 = strokes + (st * 16 + k) * 2;
        float vx = p0[0] * 256.0f, vy = p0[1] * 256.0f;
        float wx = p0[2] * 256.0f, wy = p0[3] * 256.0f;
        float sx = wx - vx, sy = wy - vy;
        float d2 = sx * sx + sy * sy;
        segA[s] = make_float4(vx, vy, sx, sy);
        segB[s] = make_float2(d2, -0.5f / (d2 + 1e-5f));
    }
    __syncthreads();

    const int  lane = tid & 31;
    const bool hiH  = lane >= 16;
    const int  m    = lane & 15;
    const int  wid  = (blockIdx.x * THREADS + tid) >> 5;  // global wave id
    const int  pid0 = wid << 5;                           // 32 pixels per wave
    const float ci  = (float)(pid0 >> 8);                 // shared row index i
    const float cj  = (float)(pid0 & 255) + 15.5f;        // shared center in j

    // Two A matrices (16x4 each), strips around the shared center:
    //   strip0: p'y = m - 15.5   strip1: p'y = m + 0.5
    // ISA layout: lanes 0-15 hold K=0,1 in v0,v1; lanes 16-31 hold K=2,3.
    float py0 = (float)m - 15.5f;
    float py1 = (float)m + 0.5f;
    v2f a0, a1;
    a0.x = hiH ? py0 * py0 : 0.0f;
    a0.y = hiH ? 1.0f      : py0;
    a1.x = hiH ? py1 * py1 : 0.0f;
    a1.y = hiH ? 1.0f      : py1;

    float minA[8], minB[8];
#pragma unroll
    for (int r = 0; r < 8; ++r) { minA[r] = 3.0e38f; minB[r] = 3.0e38f; }

    for (int g = 0; g < NGRP; ++g) {
        const int n = (g << 4) + m;
        float4 sa = segA[n];
        float2 sb = segB[n];
        // Shift segment start into the wave-centered frame (conditioning).
        float vxs  = sa.x - ci;
        float vys  = sa.y - cj;
        float pvs2 = 2.0f * (vxs * sa.z + vys * sa.w);   // +2 (v'.s)
        float vv   = vxs * vxs + vys * vys;              // |v'|^2

        // B_dot col n = (-2sx, -2sy, 0, 2 v'.s)  ->  D = -2*dot
        // B_pv2 col n = (-2v'x, -2v'y, 1, |v'|^2) -> D = |p'-v'|^2
        // ISA B layout (4x16): v0 = K=0 (low lanes)/K=2 (high), v1 = K=1/K=3.
        v2f bd, bp;
        bd.x = hiH ? 0.0f : (-2.0f * sa.z);
        bd.y = hiH ? pvs2 : (-2.0f * sa.w);
        bp.x = hiH ? 1.0f : (-2.0f * vxs);
        bp.y = hiH ? vv   : (-2.0f * vys);

        v8f c0 = {0.0f, 0.0f, 0.0f, 0.0f, 0.0f, 0.0f, 0.0f, 0.0f};
        v8f d0 = __builtin_amdgcn_wmma_f32_16x16x4_f32(
            false, a0, false, bd, (short)0, c0, false, false);
        v8f p0m = __builtin_amdgcn_wmma_f32_16x16x4_f32(
            false, a0, false, bp, (short)0, c0, false, false);
        v8f d1 = __builtin_amdgcn_wmma_f32_16x16x4_f32(
            false, a1, false, bd, (short)0, c0, false, false);
        v8f p1m = __builtin_amdgcn_wmma_f32_16x16x4_f32(
            false, a1, false, bp, (short)0, c0, false, false);

        const float d2s = sb.x, invn = sb.y;  // invn = -0.5/(d2+eps)
#pragma unroll
        for (int r = 0; r < 8; ++r) {
            // dneg = -2*dot ; t = clamp(dot/(d2+eps)) = clamp(dneg*invn)
            float dn = d0[r];
            float t  = fminf(fmaxf(dn * invn, 0.0f), 1.0f);
            float aa = fmaf(t, d2s, dn);          // t*|s|^2 - 2*dot
            minA[r]  = fminf(minA[r], fmaf(t, aa, p0m[r]));

            float dn1 = d1[r];
            float t1  = fminf(fmaxf(dn1 * invn, 0.0f), 1.0f);
            float aa1 = fmaf(t1, d2s, dn1);
            minB[r]   = fminf(minB[r], fmaf(t1, aa1, p1m[r]));
        }
    }

    // Butterfly min across each 16-lane half (cols of the tile).
#pragma unroll
    for (int r = 0; r < 8; ++r) {
        float va = minA[r], vb = minB[r];
        va = fminf(va, __shfl_xor(va, 1, 32));
        vb = fminf(vb, __shfl_xor(vb, 1, 32));
        va = fminf(va, __shfl_xor(va, 2, 32));
        vb = fminf(vb, __shfl_xor(vb, 2, 32));
        va = fminf(va, __shfl_xor(va, 4, 32));
        vb = fminf(vb, __shfl_xor(vb, 4, 32));
        va = fminf(va, __shfl_xor(va, 8, 32));
        vb = fminf(vb, __shfl_xor(vb, 8, 32));
        minA[r] = va; minB[r] = vb;
    }

    // After the butterfly every lane in a half holds the reduced value.
    // Strip0 rows 0-7 -> lane 0, rows 8-15 -> lane 16;
    // Strip1 rows 0-7 -> lane 1, rows 8-15 -> lane 17.
    if (lane < 2 || (lane >= 16 && lane < 18)) {
        const bool strip1 = (lane & 1) != 0;
        const int base = pid0 + (strip1 ? 16 : 0) + (hiH ? 8 : 0);
#pragma unroll
        for (int r = 0; r < 8; ++r) {
            float md = strip1 ? minB[r] : minA[r];
            float d = fminf(sqrtf(fmaxf(md, 0.0f)) * (1.0f / 16.0f), 1.0f);
            out[base + r] = d;
        }
    }
}

extern "C" void kernel_launch(void* const* d_in, const int* in_sizes, int n_in,
                              void* d_out, int out_size, void* d_ws, size_t ws_size,
                              hipStream_t stream) {
    (void)in_sizes; (void)n_in; (void)d_ws; (void)ws_size; (void)out_size;
    const float* strokes = (const float*)d_in[0];   // 64 x 16 x 2 fp32
    float* out = (float*)d_out;                     // 256 x 256 fp32
    // 65536 pixels / 32 per wave = 2048 waves; 8 waves (256 threads) per block.
    stroke_render_wmma<<<256, THREADS, 0, stream>>>(strokes, out);
}